// GNNBlock_48249662603742
// MI455X (gfx1250) — compile-verified
//
#include <hip/hip_runtime.h>

typedef float v2f __attribute__((ext_vector_type(2)));
typedef float v4f __attribute__((ext_vector_type(4)));
typedef float v8f __attribute__((ext_vector_type(8)));

#define N_NODES 50000
#define CCH 128
#define BN_EPS 1e-5f
#define LDA 132   // 16x128 A panel, padded stride -> bank-conflict-free frag reads

// ---------------------------------------------------------------------------
// 0) workspace init: agg = 0, deg = 1 (self loop), bn partial sums = 0
// ---------------------------------------------------------------------------
__global__ void init_ws_kernel(float* __restrict__ agg, float* __restrict__ deg,
                               float* __restrict__ bnsum, float* __restrict__ bnsumsq) {
    int idx = blockIdx.x * blockDim.x + threadIdx.x;
    const int NA = N_NODES * CCH;
    if (idx < NA) {
        agg[idx] = 0.0f;
    } else if (idx < NA + N_NODES) {
        deg[idx - NA] = 1.0f;                 // A + I degree starts at 1
    } else if (idx < NA + N_NODES + 256) {
        int j = idx - NA - N_NODES;
        if (j < 128) bnsum[j] = 0.0f; else bnsumsq[j - 128] = 0.0f;
    }
}

// ---------------------------------------------------------------------------
// 1) transpose weight [K=128][N=128] -> wT [N][K] so B-fragments are float2
// ---------------------------------------------------------------------------
__global__ void transpose_w_kernel(const float* __restrict__ w, float* __restrict__ wT) {
    int idx = blockIdx.x * blockDim.x + threadIdx.x;
    if (idx < CCH * CCH) {
        int k = idx >> 7, n = idx & 127;
        wT[n * CCH + k] = w[idx];             // coalesced read
    }
}

// ---------------------------------------------------------------------------
// 2) degree accumulation over dst
// ---------------------------------------------------------------------------
__global__ void degree_kernel(const int* __restrict__ dst, float* __restrict__ deg, int E) {
    int e = blockIdx.x * blockDim.x + threadIdx.x;
    if (e < E) atomicAdd(&deg[dst[e]], 1.0f);
}

// 3) dinv = rsqrt(deg), in place
__global__ void rsqrt_kernel(float* __restrict__ deg) {
    int i = blockIdx.x * blockDim.x + threadIdx.x;
    if (i < N_NODES) deg[i] = __frsqrt_rn(deg[i]);
}

// ---------------------------------------------------------------------------
// 4) xw = x @ W  via V_WMMA_F32_16X16X4_F32 (full f32 precision; GEMM is
//    memory-trivial at 1.6 GFLOP so f32 WMMA is the right precision choice).
//    Block = 256 threads = 8 waves; block owns a 16-row panel of x (in LDS),
//    each wave owns one 16x16 output tile (one of 8 column tiles).
// ---------------------------------------------------------------------------
__global__ void __launch_bounds__(256) gcn_gemm_wmma(const float* __restrict__ x,
                                                     const float* __restrict__ wT,
                                                     float* __restrict__ xw) {
    __shared__ float Alds[16 * LDA];

    const int tid   = threadIdx.x;
    const int tileM = blockIdx.x * 16;

    // cooperative coalesced load of the 16x128 A panel
#pragma unroll
    for (int i = 0; i < 8; ++i) {
        int idx = tid + i * 256;              // 0..2047
        int r = idx >> 7, k = idx & 127;
        Alds[r * LDA + k] = x[(tileM + r) * CCH + k];
    }
    __syncthreads();

    const int wave = tid >> 5;
    const int lane = tid & 31;
    const int half = lane >> 4;               // K sub-pair select (ISA A layout)
    const int l16  = lane & 15;               // row (A) / col (B,C)
    const int ncol = wave * 16 + l16;

    v8f acc = {0.f, 0.f, 0.f, 0.f, 0.f, 0.f, 0.f, 0.f};

#pragma unroll
    for (int kk = 0; kk < 32; ++kk) {
        const int kbase = kk * 4 + half * 2;
        // A 16x4 frag: lane<16 holds K=kbase,kbase+1 of row l16; lane>=16 K+2
        v2f a = *(const v2f*)&Alds[l16 * LDA + kbase];
        // B 4x16 frag from transposed weight: column ncol, K pair kbase
        v2f b = *(const v2f*)&wT[ncol * CCH + kbase];
        acc = __builtin_amdgcn_wmma_f32_16x16x4_f32(
            /*neg_a=*/false, a, /*neg_b=*/false, b,
            /*c_mod=*/(short)0, acc, /*reuse_a=*/false, /*reuse_b=*/false);
    }

    // C/D layout: VGPR g -> M = g (lanes 0-15) or g+8 (lanes 16-31), N = l16
#pragma unroll
    for (int g = 0; g < 8; ++g) {
        int row = tileM + g + half * 8;
        xw[row * CCH + ncol] = acc[g];
    }
}

// ---------------------------------------------------------------------------
// 5) edge scatter: agg[dst] += xw[src] * dinv[src]*dinv[dst]
//    one wave per edge; each lane handles 4 channels (float4 gather, 4 atomics)
// ---------------------------------------------------------------------------
__global__ void scatter_kernel(const int* __restrict__ src, const int* __restrict__ dst,
                               const float* __restrict__ xw, const float* __restrict__ dinv,
                               float* __restrict__ agg, int E) {
    int t = blockIdx.x * blockDim.x + threadIdx.x;
    int e = t >> 5;
    if (e >= E) return;
    int c4 = (t & 31) << 2;
    int s = src[e], d = dst[e];
    float nrm = dinv[s] * dinv[d];
    v4f v = *(const v4f*)&xw[s * CCH + c4];
    float* base = &agg[d * CCH + c4];
    atomicAdd(base + 0, v.x * nrm);
    atomicAdd(base + 1, v.y * nrm);
    atomicAdd(base + 2, v.z * nrm);
    atomicAdd(base + 3, v.w * nrm);
}

// ---------------------------------------------------------------------------
// 6) self-loop + bias, plus BN sum / sumsq (block-reduced, 128 atomics/block)
//    block covers 64 nodes x 128 channels; thread = fixed channel tid&127
// ---------------------------------------------------------------------------
__global__ void __launch_bounds__(256) selfloop_bn_kernel(float* __restrict__ agg,
                                                          const float* __restrict__ xw,
                                                          const float* __restrict__ dinv,
                                                          const float* __restrict__ bias,
                                                          float* __restrict__ bnsum,
                                                          float* __restrict__ bnsumsq) {
    __shared__ float s1[256];
    __shared__ float s2[256];
    const int c    = threadIdx.x & 127;
    const int sub  = threadIdx.x >> 7;        // 0 or 1
    const int base = blockIdx.x * 64;
    const float b  = bias[c];

    float sum = 0.f, sq = 0.f;
#pragma unroll 4
    for (int i = 0; i < 32; ++i) {
        int node = base + sub + i * 2;
        if (node < N_NODES) {
            float di  = dinv[node];
            int   off = node * CCH + c;
            float v   = agg[off] + xw[off] * (di * di) + b;
            agg[off]  = v;
            sum += v;
            sq  += v * v;
        }
    }
    s1[threadIdx.x] = sum;
    s2[threadIdx.x] = sq;
    __syncthreads();
    if (threadIdx.x < 128) {
        atomicAdd(&bnsum[c],   s1[threadIdx.x] + s1[threadIdx.x + 128]);
        atomicAdd(&bnsumsq[c], s2[threadIdx.x] + s2[threadIdx.x + 128]);
    }
}

// 7) fold BN stats + gamma/beta into per-channel scale/shift
__global__ void bn_finalize_kernel(const float* __restrict__ bnsum,
                                   const float* __restrict__ bnsumsq,
                                   const float* __restrict__ gamma,
                                   const float* __restrict__ beta,
                                   float* __restrict__ scale, float* __restrict__ shift) {
    int c = threadIdx.x;
    if (c < 128) {
        const float invN = 1.0f / (float)N_NODES;
        float mean = bnsum[c] * invN;
        float var  = bnsumsq[c] * invN - mean * mean;
        float s    = gamma[c] * __frsqrt_rn(var + BN_EPS);
        scale[c]   = s;
        shift[c]   = beta[c] - mean * s;
    }
}

// 8) normalize + ReLU
__global__ void norm_relu_kernel(const float* __restrict__ agg,
                                 const float* __restrict__ scale,
                                 const float* __restrict__ shift,
                                 float* __restrict__ out) {
    int idx = blockIdx.x * blockDim.x + threadIdx.x;
    if (idx < N_NODES * CCH) {
        int c = idx & 127;
        float v = agg[idx] * scale[c] + shift[c];
        out[idx] = v > 0.f ? v : 0.f;
    }
}

// ---------------------------------------------------------------------------
extern "C" void kernel_launch(void* const* d_in, const int* in_sizes, int n_in,
                              void* d_out, int out_size, void* d_ws, size_t ws_size,
                              hipStream_t stream) {
    const float* x      = (const float*)d_in[0];
    const int*   ei     = (const int*)d_in[1];     // [2, E]
    const float* weight = (const float*)d_in[2];
    const float* bias   = (const float*)d_in[3];
    const float* gamma  = (const float*)d_in[4];
    const float* beta   = (const float*)d_in[5];
    float*       out    = (float*)d_out;

    const int E = in_sizes[1] / 2;
    const int* src = ei;
    const int* dst = ei + E;

    // workspace layout (floats)
    float* ws      = (float*)d_ws;
    float* xw      = ws;                           // 6,400,000
    float* agg     = xw + N_NODES * CCH;           // 6,400,000
    float* deg     = agg + N_NODES * CCH;          // 50,000 (becomes dinv)
    float* wT      = deg + N_NODES;                // 16,384
    float* bnsum   = wT + CCH * CCH;               // 128
    float* bnsumsq = bnsum + 128;                  // 128
    float* scale   = bnsumsq + 128;                // 128
    float* shift   = scale + 128;                  // 128

    const int init_n = N_NODES * CCH + N_NODES + 256;
    init_ws_kernel<<<(init_n + 255) / 256, 256, 0, stream>>>(agg, deg, bnsum, bnsumsq);

    transpose_w_kernel<<<(CCH * CCH + 255) / 256, 256, 0, stream>>>(weight, wT);

    degree_kernel<<<(E + 255) / 256, 256, 0, stream>>>(dst, deg, E);
    rsqrt_kernel<<<(N_NODES + 255) / 256, 256, 0, stream>>>(deg);

    gcn_gemm_wmma<<<N_NODES / 16, 256, 0, stream>>>(x, wT, xw);   // 50000 % 16 == 0

    scatter_kernel<<<(E * 32 + 255) / 256, 256, 0, stream>>>(src, dst, xw, deg, agg, E);

    selfloop_bn_kernel<<<(N_NODES + 63) / 64, 256, 0, stream>>>(agg, xw, deg, bias,
                                                                bnsum, bnsumsq);

    bn_finalize_kernel<<<1, 128, 0, stream>>>(bnsum, bnsumsq, gamma, beta, scale, shift);

    norm_relu_kernel<<<(N_NODES * CCH + 255) / 256, 256, 0, stream>>>(agg, scale, shift, out);
}